// SpectralConv3d_41111426957953
// MI455X (gfx1250) — compile-verified
//
#include <hip/hip_runtime.h>
#include <math.h>

// ---------------------------------------------------------------------------
// SpectralConv3d on gfx1250 (MI455X): truncated 3D DFT -> per-mode complex
// channel GEMM -> truncated inverse 3D DFT.  All dense stages run on the fp32
// WMMA pipe (v_wmma_f32_16x16x4_f32); bulk global->LDS staging uses the
// CDNA5 async-to-LDS path (global_load_async_to_lds_b128 + s_wait_asynccnt).
// ---------------------------------------------------------------------------

typedef __attribute__((ext_vector_type(2))) float v2f;
typedef __attribute__((ext_vector_type(8))) float v8f;

#define PI_F 3.14159265358979323846f

// Dims
#define BB   8
#define CC   32      // in channels
#define DD   32      // out channels
#define TT   32
#define HH   96
#define WW   96
#define MT   8
#define MS   16
#define MODES (MT*MS*MS)   // 2048

__device__ __forceinline__ v8f wmma4(v2f a, v2f b, v8f c) {
  // D = A(16x4 f32) * B(4x16 f32) + C(16x16 f32)
  return __builtin_amdgcn_wmma_f32_16x16x4_f32(
      false, a, false, b, (short)0, c, false, false);
}

// Async copy of one 16-byte chunk: global -> LDS, no VGPR landing (ASYNCcnt).
__device__ __forceinline__ void async_b128(unsigned lds_byte_addr,
                                           const void* gaddr) {
  unsigned long long ga = (unsigned long long)(uintptr_t)gaddr;
  asm volatile("global_load_async_to_lds_b128 %0, %1, off"
               :: "v"(lds_byte_addr), "v"(ga)
               : "memory");
}

__device__ __forceinline__ void wait_async0() {
  asm volatile("s_wait_asynccnt 0x0" ::: "memory");
}

// Low 32 bits of an LDS generic address == LDS byte offset.
__device__ __forceinline__ unsigned lds_off(const void* p) {
  return (unsigned)(uintptr_t)p;
}

// ---------------------------------------------------------------------------
// Stage A: forward DFT along W (real -> complex, 16 modes), ortho scale here.
// One block = one (b,c,t) slab of 96x96.  6 waves, each a 16-row strip.
// GEMM: [96 x 96] * [96 x 32(cos|(-sin))]  -> real & imag 96x16.
// x slab is DMA'd to LDS asynchronously while the waves build the twiddles.
// ---------------------------------------------------------------------------
__global__ __launch_bounds__(192) void kA_dft_w(const float* __restrict__ x,
                                                float* __restrict__ xw_r,
                                                float* __restrict__ xw_i,
                                                float norm) {
  __shared__ float lx[HH * WW];      // 9216 floats
  __shared__ float bt[WW * 32];      // 3072 floats: [k][n]: n<16 cos, n>=16 -sin
  const int tid = threadIdx.x;
  const int slab = blockIdx.x;       // (b*CC+c)*TT + t
  const float* xs = x + (size_t)slab * (HH * WW);

  // Kick off async global->LDS copy of the slab: 2304 x 16B chunks,
  // 12 per thread (uniform trip count -> EXEC stays full).
  {
    const unsigned lbase = lds_off(lx);
    for (int i = tid; i < (HH * WW) / 4; i += 192)
      async_b128(lbase + (unsigned)i * 16u, xs + (size_t)i * 4);
  }
  // Overlap: build DFT coefficient matrix with VALU while DMA runs.
  for (int i = tid; i < WW * 32; i += 192) {
    int k = i / 32, n = i % 32, m = n & 15;
    float th = 2.0f * PI_F * (float)(m * k) / (float)WW;
    bt[i] = ((n < 16) ? cosf(th) : -sinf(th)) * norm;
  }
  wait_async0();
  __syncthreads();

  const int wave = tid >> 5, lane = tid & 31;
  const int m = lane & 15, hi = lane >> 4, ks = hi * 2;
  const int row = wave * 16 + m;
  v8f ar = {}, ai = {};
  for (int k0 = 0; k0 < WW; k0 += 4) {
    v2f a;  a.x  = lx[row * WW + k0 + ks];        a.y  = lx[row * WW + k0 + ks + 1];
    v2f br; br.x = bt[(k0 + ks) * 32 + m];        br.y = bt[(k0 + ks + 1) * 32 + m];
    v2f bi; bi.x = bt[(k0 + ks) * 32 + 16 + m];   bi.y = bt[(k0 + ks + 1) * 32 + 16 + m];
    ar = wmma4(a, br, ar);
    ai = wmma4(a, bi, ai);
  }
  const size_t ob = (size_t)slab * (HH * MS);
  for (int g = 0; g < 8; ++g) {
    int ro = wave * 16 + g + 8 * hi;
    xw_r[ob + ro * MS + m] = ar[g];
    xw_i[ob + ro * MS + m] = ai[g];
  }
}

// ---------------------------------------------------------------------------
// Stage B: forward DFT along H (complex, 16 modes).  E = exp(-2pi i hm h/96)
// R = cos*Xr + sin*Xi ; I = cos*Xi - sin*Xr   (Ei = -sin folded via fragments)
// One wave per (b,c,t).  8 waves / block.
// ---------------------------------------------------------------------------
__global__ __launch_bounds__(256) void kB_dft_h(const float* __restrict__ xw_r,
                                                const float* __restrict__ xw_i,
                                                float* __restrict__ xhw_r,
                                                float* __restrict__ xhw_i) {
  __shared__ float ct[MS * HH];
  __shared__ float st[MS * HH];
  const int tid = threadIdx.x;
  for (int i = tid; i < MS * HH; i += 256) {
    int mm = i / HH, h = i % HH;
    float th = 2.0f * PI_F * (float)(mm * h) / (float)HH;
    ct[i] = cosf(th); st[i] = sinf(th);
  }
  __syncthreads();
  const int wave = tid >> 5, lane = tid & 31;
  const int m = lane & 15, hi = lane >> 4, ks = hi * 2;
  const int slab = blockIdx.x * 8 + wave;   // (b*CC+c)*TT + t
  const float* xr = xw_r + (size_t)slab * (HH * MS);
  const float* xi = xw_i + (size_t)slab * (HH * MS);
  v8f aR = {}, aI = {};
  for (int k0 = 0; k0 < HH; k0 += 4) {
    v2f ac;  ac.x = ct[m * HH + k0 + ks]; ac.y = ct[m * HH + k0 + ks + 1];
    v2f as;  as.x = st[m * HH + k0 + ks]; as.y = st[m * HH + k0 + ks + 1];
    v2f ans = -as;
    v2f br; br.x = xr[(k0 + ks) * MS + m]; br.y = xr[(k0 + ks + 1) * MS + m];
    v2f bi; bi.x = xi[(k0 + ks) * MS + m]; bi.y = xi[(k0 + ks + 1) * MS + m];
    aR = wmma4(ac, br, aR);  aR = wmma4(as, bi, aR);
    aI = wmma4(ac, bi, aI);  aI = wmma4(ans, br, aI);
  }
  const size_t ob = (size_t)slab * (MS * MS);
  for (int g = 0; g < 8; ++g) {
    int hm = g + 8 * hi;
    xhw_r[ob + hm * MS + m] = aR[g];
    xhw_i[ob + hm * MS + m] = aI[g];
  }
}

// ---------------------------------------------------------------------------
// Stage C: forward DFT along T (8 modes, rows 8..15 of A are zero).
// One block per (b,c); 16 waves, one N-tile (16 of 256 spatial modes) each.
// ---------------------------------------------------------------------------
__global__ __launch_bounds__(512) void kC_dft_t(const float* __restrict__ xhw_r,
                                                const float* __restrict__ xhw_i,
                                                float* __restrict__ xf_r,
                                                float* __restrict__ xf_i) {
  __shared__ float ct[16 * TT];
  __shared__ float st[16 * TT];
  const int tid = threadIdx.x;
  for (int i = tid; i < 16 * TT; i += 512) {
    int mm = i / TT, t = i % TT;
    float th = 2.0f * PI_F * (float)(mm * t) / (float)TT;
    ct[i] = (mm < MT) ? cosf(th) : 0.0f;
    st[i] = (mm < MT) ? sinf(th) : 0.0f;
  }
  __syncthreads();
  const int wave = tid >> 5, lane = tid & 31;
  const int m = lane & 15, hi = lane >> 4, ks = hi * 2;
  const int bc = blockIdx.x;
  const int nt = wave;                       // N-tile 0..15
  const float* xr = xhw_r + (size_t)bc * TT * 256;
  const float* xi = xhw_i + (size_t)bc * TT * 256;
  v8f aR = {}, aI = {};
  for (int k0 = 0; k0 < TT; k0 += 4) {
    v2f ac;  ac.x = ct[m * TT + k0 + ks]; ac.y = ct[m * TT + k0 + ks + 1];
    v2f as;  as.x = st[m * TT + k0 + ks]; as.y = st[m * TT + k0 + ks + 1];
    v2f ans = -as;
    v2f br; br.x = xr[(size_t)(k0 + ks) * 256 + nt * 16 + m];
            br.y = xr[(size_t)(k0 + ks + 1) * 256 + nt * 16 + m];
    v2f bi; bi.x = xi[(size_t)(k0 + ks) * 256 + nt * 16 + m];
            bi.y = xi[(size_t)(k0 + ks + 1) * 256 + nt * 16 + m];
    aR = wmma4(ac, br, aR);  aR = wmma4(as, bi, aR);
    aI = wmma4(ac, bi, aI);  aI = wmma4(ans, br, aI);
  }
  if (hi == 0) {
    const size_t ob = (size_t)bc * MODES;
    for (int g = 0; g < 8; ++g) {
      xf_r[ob + g * 256 + nt * 16 + m] = aR[g];
      xf_i[ob + g * 256 + nt * 16 + m] = aI[g];
    }
  }
}

// ---------------------------------------------------------------------------
// Stage D: per-mode complex channel mixing: out[b,d] = sum_c x[b,c] * w[c,d].
// M=8 batch rows (padded to 16), K=32 channels, N=32 out channels (2 tiles).
// out_r = Xr*Wr - Xi*Wi ; out_i = Xr*Wi + Xi*Wr  (minus folded into B frag)
// One wave per mode.
// ---------------------------------------------------------------------------
__global__ __launch_bounds__(256) void kD_mix(const float* __restrict__ xf_r,
                                              const float* __restrict__ xf_i,
                                              const float* __restrict__ wr,
                                              const float* __restrict__ wi,
                                              float* __restrict__ xo_r,
                                              float* __restrict__ xo_i) {
  const int tid = threadIdx.x;
  const int wave = tid >> 5, lane = tid & 31;
  const int m = lane & 15, hi = lane >> 4, ks = hi * 2;
  const int mode = blockIdx.x * 8 + wave;     // 0..2047
  for (int dt = 0; dt < 2; ++dt) {
    v8f aR = {}, aI = {};
    for (int k0 = 0; k0 < CC; k0 += 4) {
      int c0 = k0 + ks, c1 = c0 + 1;
      v2f xar, xai;
      xar.x = (m < BB) ? xf_r[((size_t)m * CC + c0) * MODES + mode] : 0.0f;
      xar.y = (m < BB) ? xf_r[((size_t)m * CC + c1) * MODES + mode] : 0.0f;
      xai.x = (m < BB) ? xf_i[((size_t)m * CC + c0) * MODES + mode] : 0.0f;
      xai.y = (m < BB) ? xf_i[((size_t)m * CC + c1) * MODES + mode] : 0.0f;
      v2f bwr; bwr.x = wr[((size_t)c0 * DD + dt * 16 + m) * MODES + mode];
               bwr.y = wr[((size_t)c1 * DD + dt * 16 + m) * MODES + mode];
      v2f bwi; bwi.x = wi[((size_t)c0 * DD + dt * 16 + m) * MODES + mode];
               bwi.y = wi[((size_t)c1 * DD + dt * 16 + m) * MODES + mode];
      v2f bnwi = -bwi;
      aR = wmma4(xar, bwr, aR);  aR = wmma4(xai, bnwi, aR);
      aI = wmma4(xar, bwi, aI);  aI = wmma4(xai, bwr, aI);
    }
    if (hi == 0) {
      for (int g = 0; g < BB; ++g) {
        size_t oi = ((size_t)g * DD + dt * 16 + m) * MODES + mode;
        xo_r[oi] = aR[g];
        xo_i[oi] = aI[g];
      }
    }
  }
}

// ---------------------------------------------------------------------------
// Stage E: inverse DFT along T: modes(8) -> 32, exp(+2pi i tm t/32).
// y_r = cos*Xr - sin*Xi ; y_i = cos*Xi + sin*Xr.  One block per (b,d),
// 8 waves x 4 tasks covering {2 M-tiles} x {16 N-tiles}.
// ---------------------------------------------------------------------------
__global__ __launch_bounds__(256) void kE_inv_t(const float* __restrict__ xo_r,
                                                const float* __restrict__ xo_i,
                                                float* __restrict__ y1_r,
                                                float* __restrict__ y1_i) {
  __shared__ float ct[TT * MT];
  __shared__ float st[TT * MT];
  const int tid = threadIdx.x;
  for (int i = tid; i < TT * MT; i += 256) {
    int t = i / MT, k = i % MT;
    float th = 2.0f * PI_F * (float)(t * k) / (float)TT;
    ct[i] = cosf(th); st[i] = sinf(th);
  }
  __syncthreads();
  const int wave = tid >> 5, lane = tid & 31;
  const int m = lane & 15, hi = lane >> 4, ks = hi * 2;
  const int bd = blockIdx.x;
  const float* xr = xo_r + (size_t)bd * MODES;
  const float* xi = xo_i + (size_t)bd * MODES;
  for (int j = 0; j < 4; ++j) {
    int task = wave * 4 + j;
    int mt = task >> 4, nt = task & 15;
    int trow = mt * 16 + m;
    v8f aR = {}, aI = {};
    for (int k0 = 0; k0 < MT; k0 += 4) {
      v2f ac;  ac.x = ct[trow * MT + k0 + ks]; ac.y = ct[trow * MT + k0 + ks + 1];
      v2f as;  as.x = st[trow * MT + k0 + ks]; as.y = st[trow * MT + k0 + ks + 1];
      v2f ans = -as;
      v2f br; br.x = xr[(k0 + ks) * 256 + nt * 16 + m];
              br.y = xr[(k0 + ks + 1) * 256 + nt * 16 + m];
      v2f bi; bi.x = xi[(k0 + ks) * 256 + nt * 16 + m];
              bi.y = xi[(k0 + ks + 1) * 256 + nt * 16 + m];
      aR = wmma4(ac, br, aR);  aR = wmma4(ans, bi, aR);
      aI = wmma4(ac, bi, aI);  aI = wmma4(as, br, aI);
    }
    const size_t ob = (size_t)bd * TT * 256;
    for (int g = 0; g < 8; ++g) {
      int ro = mt * 16 + g + 8 * hi;
      y1_r[ob + (size_t)ro * 256 + nt * 16 + m] = aR[g];
      y1_i[ob + (size_t)ro * 256 + nt * 16 + m] = aI[g];
    }
  }
}

// ---------------------------------------------------------------------------
// Stage F: inverse DFT along H: modes(16) -> 96.  One block per (b,d,t),
// 6 waves, each a 16-row strip of H.
// ---------------------------------------------------------------------------
__global__ __launch_bounds__(192) void kF_inv_h(const float* __restrict__ y1_r,
                                                const float* __restrict__ y1_i,
                                                float* __restrict__ y2_r,
                                                float* __restrict__ y2_i) {
  __shared__ float ct[HH * MS];
  __shared__ float st[HH * MS];
  const int tid = threadIdx.x;
  for (int i = tid; i < HH * MS; i += 192) {
    int h = i / MS, k = i % MS;
    float th = 2.0f * PI_F * (float)(h * k) / (float)HH;
    ct[i] = cosf(th); st[i] = sinf(th);
  }
  __syncthreads();
  const int wave = tid >> 5, lane = tid & 31;
  const int m = lane & 15, hi = lane >> 4, ks = hi * 2;
  const int slab = blockIdx.x;               // (b*DD+d)*TT + t
  const float* xr = y1_r + (size_t)slab * 256;
  const float* xi = y1_i + (size_t)slab * 256;
  const int hrow = wave * 16 + m;
  v8f aR = {}, aI = {};
  for (int k0 = 0; k0 < MS; k0 += 4) {
    v2f ac;  ac.x = ct[hrow * MS + k0 + ks]; ac.y = ct[hrow * MS + k0 + ks + 1];
    v2f as;  as.x = st[hrow * MS + k0 + ks]; as.y = st[hrow * MS + k0 + ks + 1];
    v2f ans = -as;
    v2f br; br.x = xr[(k0 + ks) * MS + m]; br.y = xr[(k0 + ks + 1) * MS + m];
    v2f bi; bi.x = xi[(k0 + ks) * MS + m]; bi.y = xi[(k0 + ks + 1) * MS + m];
    aR = wmma4(ac, br, aR);  aR = wmma4(ans, bi, aR);
    aI = wmma4(ac, bi, aI);  aI = wmma4(as, br, aI);
  }
  const size_t ob = (size_t)slab * (HH * MS);
  for (int g = 0; g < 8; ++g) {
    int ro = wave * 16 + g + 8 * hi;
    y2_r[ob + ro * MS + m] = aR[g];
    y2_i[ob + ro * MS + m] = aI[g];
  }
}

// ---------------------------------------------------------------------------
// Stage G: inverse rfft along W: 16 complex modes -> 96 real points.
// Hermitian reconstruction folded into coeff matrix: k=0 -> (1, 0),
// k>0 -> (2cos, -2sin).  K packed as [Y2r(0..15) | Y2i(0..15)].
// One block per (b,d,t) slab; 6 waves x 6 N-tiles; writes d_out.
// y2 planes are DMA'd to LDS asynchronously while the coeff table is built.
// ---------------------------------------------------------------------------
__global__ __launch_bounds__(192) void kG_inv_w(const float* __restrict__ y2_r,
                                                const float* __restrict__ y2_i,
                                                float* __restrict__ out,
                                                float norm) {
  __shared__ float ctab[32 * WW];   // [kk][w]
  __shared__ float lyr[HH * MS];
  __shared__ float lyi[HH * MS];
  const int tid = threadIdx.x;
  const int slab = blockIdx.x;      // (b*DD+d)*TT + t

  // Async global->LDS staging of both complex planes (384 chunks each; 2
  // uniform iterations per thread per plane -> EXEC stays full).
  {
    const unsigned lr = lds_off(lyr), li = lds_off(lyi);
    const float* gr = y2_r + (size_t)slab * (HH * MS);
    const float* gi = y2_i + (size_t)slab * (HH * MS);
    for (int i = tid; i < (HH * MS) / 4; i += 192) {
      async_b128(lr + (unsigned)i * 16u, gr + (size_t)i * 4);
      async_b128(li + (unsigned)i * 16u, gi + (size_t)i * 4);
    }
  }
  // Overlap: build the irfft coefficient matrix while the DMA runs.
  for (int i = tid; i < 32 * WW; i += 192) {
    int kk = i / WW, w = i % WW, k = kk & 15;
    float th = 2.0f * PI_F * (float)(k * w) / (float)WW;
    float v;
    if (kk < 16) v = (k == 0) ? 1.0f : 2.0f * cosf(th);
    else         v = (k == 0) ? 0.0f : -2.0f * sinf(th);
    ctab[i] = v * norm;
  }
  wait_async0();
  __syncthreads();

  const int wave = tid >> 5, lane = tid & 31;
  const int m = lane & 15, hi = lane >> 4, ks = hi * 2;
  const int hrow = wave * 16 + m;
  for (int nt = 0; nt < 6; ++nt) {
    v8f acc = {};
    for (int k0 = 0; k0 < 32; k0 += 4) {
      int kk0 = k0 + ks, kk1 = kk0 + 1;
      v2f a;
      a.x = (kk0 < 16) ? lyr[hrow * MS + kk0] : lyi[hrow * MS + kk0 - 16];
      a.y = (kk1 < 16) ? lyr[hrow * MS + kk1] : lyi[hrow * MS + kk1 - 16];
      v2f b;
      b.x = ctab[kk0 * WW + nt * 16 + m];
      b.y = ctab[kk1 * WW + nt * 16 + m];
      acc = wmma4(a, b, acc);
    }
    const size_t ob = (size_t)slab * (HH * WW);
    for (int g = 0; g < 8; ++g) {
      int ro = wave * 16 + g + 8 * hi;
      out[ob + (size_t)ro * WW + nt * 16 + m] = acc[g];
    }
  }
}

// ---------------------------------------------------------------------------
// Launch
// ---------------------------------------------------------------------------
extern "C" void kernel_launch(void* const* d_in, const int* in_sizes, int n_in,
                              void* d_out, int out_size, void* d_ws, size_t ws_size,
                              hipStream_t stream) {
  const float* x  = (const float*)d_in[0];
  const float* wr = (const float*)d_in[1];
  const float* wi = (const float*)d_in[2];
  float* out = (float*)d_out;
  float* ws  = (float*)d_ws;

  // Workspace layout (floats).  Inverse stages reuse forward buffers.
  const size_t XW_SZ  = (size_t)BB * CC * TT * HH * MS;   // 12,582,912
  const size_t XHW_SZ = (size_t)BB * CC * TT * MS * MS;   //  2,097,152
  const size_t XF_SZ  = (size_t)BB * CC * MODES;          //    524,288
  float* xw_r  = ws;                 float* xw_i  = xw_r  + XW_SZ;
  float* xhw_r = xw_i  + XW_SZ;      float* xhw_i = xhw_r + XHW_SZ;
  float* xf_r  = xhw_i + XHW_SZ;     float* xf_i  = xf_r  + XF_SZ;
  float* xo_r  = xf_i  + XF_SZ;      float* xo_i  = xo_r  + XF_SZ;
  float* y1_r  = xhw_r;              float* y1_i  = xhw_i;   // reuse
  float* y2_r  = xw_r;               float* y2_i  = xw_i;    // reuse

  const float norm = 1.0f / sqrtf((float)(TT * HH * WW));   // ortho (one side)

  kA_dft_w<<<BB * CC * TT, 192, 0, stream>>>(x, xw_r, xw_i, norm);
  kB_dft_h<<<(BB * CC * TT) / 8, 256, 0, stream>>>(xw_r, xw_i, xhw_r, xhw_i);
  kC_dft_t<<<BB * CC, 512, 0, stream>>>(xhw_r, xhw_i, xf_r, xf_i);
  kD_mix  <<<MODES / 8, 256, 0, stream>>>(xf_r, xf_i, wr, wi, xo_r, xo_i);
  kE_inv_t<<<BB * DD, 256, 0, stream>>>(xo_r, xo_i, y1_r, y1_i);
  kF_inv_h<<<BB * DD * TT, 192, 0, stream>>>(y1_r, y1_i, y2_r, y2_i);
  kG_inv_w<<<BB * DD * TT, 192, 0, stream>>>(y2_r, y2_i, out, norm);
}